// InteractionNetwork_90469191123233
// MI455X (gfx1250) — compile-verified
//
#include <hip/hip_runtime.h>
#include <hip/hip_bf16.h>

typedef float v2f __attribute__((ext_vector_type(2)));
typedef float v8f __attribute__((ext_vector_type(8)));

#define N_NODES     50000
#define N_EDGES     800000
#define D_S         64
#define D_R         16
#define D_E         64
#define D_X         16
#define D_P         64
#define NUM_CLASSES 10
#define KTOT        144     // 2*D_S + D_R == D_S + D_X + D_E
#define AST         148     // padded LDS row stride (floats): 16B-aligned rows, good bank spread
#define PST         65      // padded P-tile stride
#define WSZ         (KTOT * D_E)   // 9216 floats per packed weight matrix

// ---------------------------------------------------------------------------
// Pre-pass: pack W_r / W_o into WMMA B-fragment order so each lane's
// {W[k0][n], W[k0+1][n]} pair is one contiguous b64 load.
//   dst = (((kc*2 + khalf)*4 + nt)*16 + nn)*2 + comp
// with k = 4*kc + 2*khalf + comp, n = 16*nt + nn.
// ---------------------------------------------------------------------------
__global__ void pack_w_kernel(const float* __restrict__ Wr,
                              const float* __restrict__ Wo,
                              float* __restrict__ WrP, float* __restrict__ WoP)
{
    int i = blockIdx.x * 256 + threadIdx.x;      // i = k*64 + n
    if (i >= WSZ) return;
    int k = i >> 6, n = i & 63;
    int kc = k >> 2, rem = k & 3;
    int khalf = rem >> 1, comp = rem & 1;
    int nt = n >> 4, nn = n & 15;
    int dst = (((kc * 2 + khalf) * 4 + nt) * 16 + nn) * 2 + comp;
    WrP[dst] = Wr[i];
    WoP[dst] = Wo[i];
}

// ---------------------------------------------------------------------------
// Kernel A: edge GEMM [64 edges x 144] @ W_r[144 x 64] via
// V_WMMA_F32_16X16X4_F32 (4 M-tiles x 4 N-tiles of accumulators per wave,
// B fragments reused across the 4 M-tiles), then atomic scatter into E_hat.
// One wave per block; 800000/64 = 12500 blocks exactly.
// ---------------------------------------------------------------------------
__global__ __launch_bounds__(32) void in_edge_kernel(
    const float* __restrict__ O,  const float* __restrict__ Ra,
    const int*   __restrict__ snd, const int*  __restrict__ rcv,
    const float* __restrict__ WrP, const float* __restrict__ br,
    float* __restrict__ Ehat)
{
    __shared__ float As[64 * AST];               // 37,888 B
    const int lane  = threadIdx.x;               // 0..31
    const int ebase = blockIdx.x * 64;

    // Stage A-tile (64 edges x 144): [0,64)=O[snd], [64,128)=O[rcv], [128,144)=R_a
    for (int idx = lane; idx < 64 * 16; idx += 32) {          // float4 granules
        int m = idx >> 4, q = (idx & 15) << 2;
        int s = snd[ebase + m];
        *(float4*)(As + m * AST + q) = *(const float4*)(O + (size_t)s * D_S + q);
    }
    for (int idx = lane; idx < 64 * 16; idx += 32) {
        int m = idx >> 4, q = (idx & 15) << 2;
        int r = rcv[ebase + m];
        *(float4*)(As + m * AST + 64 + q) = *(const float4*)(O + (size_t)r * D_S + q);
    }
    for (int idx = lane; idx < 64 * 4; idx += 32) {
        int m = idx >> 2, q = (idx & 3) << 2;
        *(float4*)(As + m * AST + 128 + q) = *(const float4*)(Ra + (size_t)(ebase + m) * D_R + q);
    }
    __syncthreads();

    // Fragment indexing (ISA 7.12.2, 32-bit):
    //   A(m,k): lane = m + 16*(k>>1), vgpr = k&1
    //   B(k,n): lane = n + 16*(k>>1), vgpr = k&1
    //   D(m,n): lane = n + 16*(m>>3), vgpr = m&7
    const int half = lane >> 4;
    const int mn   = lane & 15;
    v8f acc[4][4] = {};                          // [m-tile][n-tile]

    for (int kc = 0; kc < KTOT / 4; ++kc) {
        const int k0 = kc * 4 + half * 2;
        const float* wp = WrP + (size_t)((kc * 2 + half) * 4) * 32 + mn * 2;
        v2f b0 = *(const v2f*)(wp);
        v2f b1 = *(const v2f*)(wp + 32);
        v2f b2 = *(const v2f*)(wp + 64);
        v2f b3 = *(const v2f*)(wp + 96);
#pragma unroll
        for (int t = 0; t < 4; ++t) {
            v2f a = *(const v2f*)(As + (t * 16 + mn) * AST + k0);
            acc[t][0] = __builtin_amdgcn_wmma_f32_16x16x4_f32(false, a, false, b0, (short)0, acc[t][0], false, false);
            acc[t][1] = __builtin_amdgcn_wmma_f32_16x16x4_f32(false, a, false, b1, (short)0, acc[t][1], false, false);
            acc[t][2] = __builtin_amdgcn_wmma_f32_16x16x4_f32(false, a, false, b2, (short)0, acc[t][2], false, false);
            acc[t][3] = __builtin_amdgcn_wmma_f32_16x16x4_f32(false, a, false, b3, (short)0, acc[t][3], false, false);
        }
    }

    // Bias + segment-sum scatter: lane holds rows m = 16*t + 8*half + j, col mn.
    const float bias0 = br[mn];
    const float bias1 = br[16 + mn];
    const float bias2 = br[32 + mn];
    const float bias3 = br[48 + mn];
#pragma unroll
    for (int t = 0; t < 4; ++t) {
#pragma unroll
        for (int j = 0; j < 8; ++j) {
            int r = rcv[ebase + t * 16 + half * 8 + j];
            size_t rb = (size_t)r * D_E;
            atomicAdd(Ehat + rb      + mn, acc[t][0][j] + bias0);
            atomicAdd(Ehat + rb + 16 + mn, acc[t][1][j] + bias1);
            atomicAdd(Ehat + rb + 32 + mn, acc[t][2][j] + bias2);
            atomicAdd(Ehat + rb + 48 + mn, acc[t][3][j] + bias3);
        }
    }
}

// ---------------------------------------------------------------------------
// Kernel B: node GEMM [16 nodes x 144] @ W_o[144 x 64] via WMMA (packed W_o),
// then fused P @ W_s[64 x 10] + softmax. 3125 tiles = 625 blocks x 5 waves.
// ---------------------------------------------------------------------------
__global__ __launch_bounds__(160) void in_node_kernel(
    const float* __restrict__ O,  const float* __restrict__ X,
    const float* __restrict__ Ehat,
    const float* __restrict__ WoP, const float* __restrict__ bo,
    const float* __restrict__ Ws, const float* __restrict__ bs,
    float* __restrict__ out)
{
    __shared__ float smem[5][16 * AST];
    const int wave  = threadIdx.x >> 5;
    const int lane  = threadIdx.x & 31;
    const int tile  = blockIdx.x * 5 + wave;     // 0 .. 3124
    const int nodeb = tile << 4;
    float* As = smem[wave];

    // Stage C-tile: [0,64)=O, [64,80)=X, [80,144)=E_hat
    for (int idx = lane; idx < 16 * 16; idx += 32) {
        int m = idx >> 4, q = (idx & 15) << 2;
        *(float4*)(As + m * AST + q) = *(const float4*)(O + (size_t)(nodeb + m) * D_S + q);
    }
    for (int idx = lane; idx < 16 * 4; idx += 32) {
        int m = idx >> 2, q = (idx & 3) << 2;
        *(float4*)(As + m * AST + 64 + q) = *(const float4*)(X + (size_t)(nodeb + m) * D_X + q);
    }
    for (int idx = lane; idx < 16 * 16; idx += 32) {
        int m = idx >> 4, q = (idx & 15) << 2;
        *(float4*)(As + m * AST + 80 + q) = *(const float4*)(Ehat + (size_t)(nodeb + m) * D_E + q);
    }
    __syncthreads();

    const int half = lane >> 4;
    const int mn   = lane & 15;
    v8f acc0 = {}, acc1 = {}, acc2 = {}, acc3 = {};

    for (int kc = 0; kc < KTOT / 4; ++kc) {
        const int k0 = kc * 4 + half * 2;
        v2f a = *(const v2f*)(As + mn * AST + k0);
        const float* wp = WoP + (size_t)((kc * 2 + half) * 4) * 32 + mn * 2;
        v2f b0 = *(const v2f*)(wp);
        v2f b1 = *(const v2f*)(wp + 32);
        v2f b2 = *(const v2f*)(wp + 64);
        v2f b3 = *(const v2f*)(wp + 96);
        acc0 = __builtin_amdgcn_wmma_f32_16x16x4_f32(false, a, false, b0, (short)0, acc0, false, false);
        acc1 = __builtin_amdgcn_wmma_f32_16x16x4_f32(false, a, false, b1, (short)0, acc1, false, false);
        acc2 = __builtin_amdgcn_wmma_f32_16x16x4_f32(false, a, false, b2, (short)0, acc2, false, false);
        acc3 = __builtin_amdgcn_wmma_f32_16x16x4_f32(false, a, false, b3, (short)0, acc3, false, false);
    }

    // P-tile (with bias) back to LDS (reuse A region), then per-node head.
    float* Ps = smem[wave];
    const float bo0 = bo[mn], bo1 = bo[16 + mn], bo2 = bo[32 + mn], bo3 = bo[48 + mn];
    const int mbase = half * 8;
#pragma unroll
    for (int j = 0; j < 8; ++j) {
        int m = mbase + j;
        Ps[m * PST      + mn] = acc0[j] + bo0;
        Ps[m * PST + 16 + mn] = acc1[j] + bo1;
        Ps[m * PST + 32 + mn] = acc2[j] + bo2;
        Ps[m * PST + 48 + mn] = acc3[j] + bo3;
    }
    __syncthreads();

    if (lane < 16) {
        const int node = nodeb + lane;
        float sc[NUM_CLASSES];
#pragma unroll
        for (int c = 0; c < NUM_CLASSES; ++c) sc[c] = bs[c];
        for (int k = 0; k < D_P; ++k) {
            float p = Ps[lane * PST + k];
#pragma unroll
            for (int c = 0; c < NUM_CLASSES; ++c) sc[c] = fmaf(p, Ws[k * NUM_CLASSES + c], sc[c]);
        }
        float mx = sc[0];
#pragma unroll
        for (int c = 1; c < NUM_CLASSES; ++c) mx = fmaxf(mx, sc[c]);
        float sum = 0.f;
#pragma unroll
        for (int c = 0; c < NUM_CLASSES; ++c) { sc[c] = expf(sc[c] - mx); sum += sc[c]; }
        const float inv = 1.f / sum;
#pragma unroll
        for (int c = 0; c < NUM_CLASSES; ++c)
            out[(size_t)node * NUM_CLASSES + c] = sc[c] * inv;
    }
}

extern "C" void kernel_launch(void* const* d_in, const int* in_sizes, int n_in,
                              void* d_out, int out_size, void* d_ws, size_t ws_size,
                              hipStream_t stream) {
    const float* O  = (const float*)d_in[0];
    const float* X  = (const float*)d_in[1];
    const float* Ra = (const float*)d_in[2];
    const int*   sn = (const int*)  d_in[3];
    const int*   rc = (const int*)  d_in[4];
    const float* Wr = (const float*)d_in[5];
    const float* br = (const float*)d_in[6];
    const float* Wo = (const float*)d_in[7];
    const float* bo = (const float*)d_in[8];
    const float* Ws = (const float*)d_in[9];
    const float* bs = (const float*)d_in[10];
    float* out  = (float*)d_out;

    float* Ehat = (float*)d_ws;                  // N_NODES*D_E floats (12.8 MB)
    float* WrP  = Ehat + (size_t)N_NODES * D_E;  // 9216 floats, 8B-aligned
    float* WoP  = WrP + WSZ;                     // 9216 floats

    hipMemsetAsync(Ehat, 0, (size_t)N_NODES * D_E * sizeof(float), stream);
    pack_w_kernel<<<(WSZ + 255) / 256, 256, 0, stream>>>(Wr, Wo, WrP, WoP);

    // 800000 edges / 64 per wave = 12500 single-wave blocks (exact)
    in_edge_kernel<<<N_EDGES / 64, 32, 0, stream>>>(O, Ra, sn, rc, WrP, br, Ehat);
    // 50000 nodes / 16 per wave / 5 waves per block = 625 blocks (exact)
    in_node_kernel<<<N_NODES / 16 / 5, 160, 0, stream>>>(O, X, Ehat, WoP, bo, Ws, bs, out);
}